// MPS_TDVP_79577154060544
// MI455X (gfx1250) — compile-verified
//
#include <hip/hip_runtime.h>
#include <math.h>

// MPS-TDVP local-Hamiltonian kernel for MI455X (gfx1250, wave32).
//
//  K1: per (chunk of 16 sites, 16-batch tile): build per-site 8x8 transfer
//      matrices A via V_WMMA_F32_16X16X4_F32 (A[b,(j,l)] = y[b,(i,k)] @ W),
//      chain them into a chunk product P_c (row-parallel VALU), store to ws.
//  K2: per batch: 64-step sequential boundary scan over chunk products ->
//      left row-vectors / right col-vectors at every chunk boundary.
//  K3: per (chunk, batch tile): recompute A via WMMA, propagate left/right
//      vectors through the chunk in LDS, form H, apply to raw x, activation,
//      L1-normalize, write out.
//
// gfx1250 specifics used: v_wmma_f32_16x16x4_f32 for the transfer-matrix
// contraction (K=4 exactly matches d*d), global_load_async_to_lds_b128 +
// s_wait_asynccnt for MPO staging into LDS, wave32 block sizing (4 waves),
// double-buffered LDS so each serial chain step costs one split barrier.
//
// Workspace use (d_ws): 64*512*64 + 2*64*512*8 floats = 10 MB.

#define NSITES 1024
#define NBATCH 512
#define DB     8
#define CHUNK  16
#define NCHUNK (NSITES / CHUNK)   // 64
#define BTILE  16

typedef float v2f __attribute__((ext_vector_type(2)));
typedef float v8f __attribute__((ext_vector_type(8)));
typedef int b128_t __attribute__((vector_size(4 * sizeof(int))));

#if defined(__HIP_DEVICE_COMPILE__) && __has_builtin(__builtin_amdgcn_wmma_f32_16x16x4_f32)
#define HAVE_WMMA_F32_K4 1
#else
#define HAVE_WMMA_F32_K4 0
#endif

#if defined(__HIP_DEVICE_COMPILE__) &&                                  \
    __has_builtin(__builtin_amdgcn_global_load_async_to_lds_b128) &&    \
    __has_builtin(__builtin_amdgcn_s_wait_asynccnt)
#define HAVE_ASYNC_LDS 1
#else
#define HAVE_ASYNC_LDS 0
#endif

#define GAS __attribute__((address_space(1)))
#define LAS __attribute__((address_space(3)))

// Global -> LDS copy of nfloats (multiple of 4) floats. On gfx1250 uses the
// async-DMA path (ASYNCcnt); caller must call g2l_wait() then __syncthreads().
__device__ __forceinline__ void g2l_copy(const float* __restrict__ g,
                                         float* __restrict__ l, int nfloats,
                                         int t, int nthreads)
{
#if HAVE_ASYNC_LDS
  for (int i = t * 4; i < nfloats; i += nthreads * 4)
    __builtin_amdgcn_global_load_async_to_lds_b128((GAS b128_t*)(g + i),
                                                   (LAS b128_t*)(l + i), 0, 0);
#else
  for (int i = t; i < nfloats; i += nthreads) l[i] = g[i];
#endif
}

__device__ __forceinline__ void g2l_wait()
{
#if HAVE_ASYNC_LDS
  __builtin_amdgcn_s_wait_asynccnt(0);
#endif
}

// Compute one 16-column (j,l)-tile of the per-site transfer matrix
//   A[b, j*8+l] = sum_{i,k in {0,1}} xn[b,i]*xn[b,k] * W[j,i,k,l]
// for a 16-batch tile, as a 16x16x4 f32 WMMA:  D(16x16) = Y(16x4) @ M(4x16).
// Must be executed by a full wave (EXEC all ones). Writes into sA[b*64 + jl].
//
// WMMA operand layout (CDNA5 ISA 7.12.2, f32 16x4 A-matrix / 16x16 C/D):
//   A: lane = 16*kh + m holds row M=m, K = {2*kh, 2*kh+1} in vgpr {0,1}
//   B: mirror of A: lane = 16*kh + n holds col N=n, K = {2*kh, 2*kh+1}
//   D: vgpr r: lanes 0-15 -> (M=r, N=lane), lanes 16-31 -> (M=r+8, N=lane-16)
__device__ __forceinline__ void site_transfer_tile(
    const float* __restrict__ Wsite,  // 256 floats: j*32 + i*16 + k*8 + l
    const float* __restrict__ xnb,    // 16 batches x 2: [b*2 + p]
    float* __restrict__ sA,           // 16 x 64 out: [b*64 + jl]
    int tile, int lane)
{
  const int m  = lane & 15;   // batch row (A) / output column within tile (B/D)
  const int hi = lane >> 4;   // K-half select
#if HAVE_WMMA_F32_K4
  v2f a;
  {
    const float xi = xnb[m * 2 + hi];          // i = hi
    a.x = xi * xnb[m * 2 + 0];                 // K = 2*hi + 0  (k = 0)
    a.y = xi * xnb[m * 2 + 1];                 // K = 2*hi + 1  (k = 1)
  }
  v2f b;
  {
    const int jl = tile * 16 + m;
    const int j = jl >> 3, l = jl & 7;
    b.x = Wsite[j * 32 + hi * 16 + 0 * 8 + l]; // K = 2*hi + 0
    b.y = Wsite[j * 32 + hi * 16 + 1 * 8 + l]; // K = 2*hi + 1
  }
  v8f c = {0.f, 0.f, 0.f, 0.f, 0.f, 0.f, 0.f, 0.f};
  c = __builtin_amdgcn_wmma_f32_16x16x4_f32(false, a, false, b, (short)0, c,
                                            false, false);
  const int brow = hi * 8;     // D-row (batch) base for this lane half
#pragma unroll
  for (int r = 0; r < 8; ++r)
    sA[(brow + r) * 64 + tile * 16 + m] = c[r];
#else
  // Scalar fallback: lane computes 8 outputs for batch m, columns hi*8..hi*8+7.
  for (int cc = 0; cc < 8; ++cc) {
    const int col = tile * 16 + hi * 8 + cc;
    const int j = col >> 3, l = col & 7;
    float acc = 0.f;
    for (int K = 0; K < 4; ++K) {
      const float y = xnb[m * 2 + (K >> 1)] * xnb[m * 2 + (K & 1)];
      acc += y * Wsite[j * 32 + (K >> 1) * 16 + (K & 1) * 8 + l];
    }
    sA[m * 64 + col] = acc;
  }
#endif
}

// ---------------------------------------------------------------------------
// K1: per-site transfer matrices (WMMA) + chunk products.
// grid = (NCHUNK, NBATCH/BTILE), block = 128 (4 waves).
// ---------------------------------------------------------------------------
__global__ __launch_bounds__(128) void k_chunk_transfer(
    const float* __restrict__ x, const float* __restrict__ mpo,
    float* __restrict__ P)
{
  const int c = blockIdx.x, bt = blockIdx.y, t = threadIdx.x;
  const int lane = t & 31, wave = t >> 5;

  __shared__ float sW[CHUNK * 256];        // 16 KB: mpo sites of this chunk
  __shared__ float sXn[CHUNK * BTILE * 2]; //  2 KB: normalized x
  __shared__ float sA[2][BTILE * 64];      //  8 KB: double-buffered A

  g2l_copy(mpo + (size_t)c * CHUNK * 256, sW, CHUNK * 256, t, 128);
  for (int p = t; p < CHUNK * BTILE; p += 128) {
    const int s = p / BTILE, b = p % BTILE;
    const size_t base = ((size_t)(c * CHUNK + s) * NBATCH + bt * BTILE + b) * 2;
    const float x0 = x[base + 0], x1 = x[base + 1];
    const float rn = 1.0f / sqrtf(x0 * x0 + x1 * x1);
    sXn[p * 2 + 0] = x0 * rn;
    sXn[p * 2 + 1] = x1 * rn;
  }
  g2l_wait();
  __syncthreads();

  // Chunk product P_c = A_0 @ A_1 @ ... @ A_15, row-parallel:
  // thread (b = t>>3, j = t&7) carries row j of the running product.
  const int rb = t >> 3, rj = t & 7;
  float row[8];

  for (int s = 0; s < CHUNK; ++s) {
    float* buf = sA[s & 1];
    site_transfer_tile(&sW[s * 256], &sXn[s * BTILE * 2], buf, wave, lane);
    __syncthreads();
    if (s == 0) {
#pragma unroll
      for (int l = 0; l < 8; ++l) row[l] = buf[rb * 64 + rj * 8 + l];
    } else {
      float nrow[8];
#pragma unroll
      for (int l = 0; l < 8; ++l) {
        float acc = 0.f;
#pragma unroll
        for (int mm = 0; mm < 8; ++mm)
          acc += row[mm] * buf[rb * 64 + mm * 8 + l];
        nrow[l] = acc;
      }
#pragma unroll
      for (int l = 0; l < 8; ++l) row[l] = nrow[l];
    }
  }
  float* Pout = P + (((size_t)c * NBATCH + bt * BTILE + rb) * 8 + rj) * 8;
#pragma unroll
  for (int l = 0; l < 8; ++l) Pout[l] = row[l];
}

// ---------------------------------------------------------------------------
// K2: 64-step boundary scan per batch (left rows forward, right cols backward).
// ---------------------------------------------------------------------------
__global__ __launch_bounds__(128) void k_boundary(
    const float* __restrict__ P, float* __restrict__ Lrow,
    float* __restrict__ Rcol)
{
  const int tid = blockIdx.x * 128 + threadIdx.x;
  if (tid >= 2 * NBATCH) return;
  const int b = tid & (NBATCH - 1);
  float v[8];
#pragma unroll
  for (int i = 0; i < 8; ++i) v[i] = 0.f;
  v[0] = 1.f;

  if (tid < NBATCH) {                       // Lrow[c] = e0^T P_0 ... P_{c-1}
#pragma unroll
    for (int i = 0; i < 8; ++i) Lrow[(size_t)b * 8 + i] = v[i];
    for (int c = 0; c < NCHUNK - 1; ++c) {
      const float* Pc = P + ((size_t)c * NBATCH + b) * 64;
      float nv[8];
#pragma unroll
      for (int l = 0; l < 8; ++l) {
        float a = 0.f;
#pragma unroll
        for (int j = 0; j < 8; ++j) a += v[j] * Pc[j * 8 + l];
        nv[l] = a;
      }
#pragma unroll
      for (int l = 0; l < 8; ++l) {
        v[l] = nv[l];
        Lrow[((size_t)(c + 1) * NBATCH + b) * 8 + l] = nv[l];
      }
    }
  } else {                                  // Rcol[c] = P_{c+1} ... P_{63} e0
#pragma unroll
    for (int i = 0; i < 8; ++i)
      Rcol[((size_t)(NCHUNK - 1) * NBATCH + b) * 8 + i] = v[i];
    for (int c = NCHUNK - 1; c >= 1; --c) {
      const float* Pc = P + ((size_t)c * NBATCH + b) * 64;
      float nv[8];
#pragma unroll
      for (int j = 0; j < 8; ++j) {
        float a = 0.f;
#pragma unroll
        for (int l = 0; l < 8; ++l) a += Pc[j * 8 + l] * v[l];
        nv[j] = a;
      }
#pragma unroll
      for (int j = 0; j < 8; ++j) {
        v[j] = nv[j];
        Rcol[((size_t)(c - 1) * NBATCH + b) * 8 + j] = nv[j];
      }
    }
  }
}

// ---------------------------------------------------------------------------
// K3: recompute A (WMMA), in-chunk left/right propagation, local H, output.
// grid = (NCHUNK, NBATCH/BTILE), block = 128.
// ---------------------------------------------------------------------------
__global__ __launch_bounds__(128) void k_local(
    const float* __restrict__ x, const float* __restrict__ mpo,
    const float* __restrict__ Lrow, const float* __restrict__ Rcol,
    float* __restrict__ out)
{
  const int c = blockIdx.x, bt = blockIdx.y, t = threadIdx.x;
  const int lane = t & 31, wave = t >> 5;

  __shared__ float sW[CHUNK * 256];          // 16 KB
  __shared__ float sXn[CHUNK * BTILE * 2];   //  2 KB normalized x
  __shared__ float sXr[CHUNK * BTILE * 2];   //  2 KB raw x
  __shared__ float sA[2][BTILE * 64];        //  8 KB double-buffered A
  __shared__ float sVl[CHUNK][BTILE][8];     //  8 KB Aleft per site
  __shared__ float sVr[CHUNK][BTILE][8];     //  8 KB Aright per site
  __shared__ float sv[BTILE][8];
  __shared__ float sw[BTILE][8];

  g2l_copy(mpo + (size_t)c * CHUNK * 256, sW, CHUNK * 256, t, 128);
  for (int p = t; p < CHUNK * BTILE; p += 128) {
    const int s = p / BTILE, b = p % BTILE;
    const size_t base = ((size_t)(c * CHUNK + s) * NBATCH + bt * BTILE + b) * 2;
    const float x0 = x[base + 0], x1 = x[base + 1];
    sXr[p * 2 + 0] = x0;
    sXr[p * 2 + 1] = x1;
    const float rn = 1.0f / sqrtf(x0 * x0 + x1 * x1);
    sXn[p * 2 + 0] = x0 * rn;
    sXn[p * 2 + 1] = x1 * rn;
  }
  const int rb = t >> 3, re = t & 7;
  sv[rb][re] = Lrow[((size_t)c * NBATCH + bt * BTILE + rb) * 8 + re];
  sw[rb][re] = Rcol[((size_t)c * NBATCH + bt * BTILE + rb) * 8 + re];
  g2l_wait();
  __syncthreads();

  // Forward: Aleft[site s] = sv before applying A_s ; sv <- sv @ A_s.
  // (The 8 threads of each batch group live in one wave; their LDS ops on
  //  sv[rb][*] execute in wave program order, and the per-site barrier orders
  //  everything across iterations.)
  for (int s = 0; s < CHUNK; ++s) {
    float* buf = sA[s & 1];
    site_transfer_tile(&sW[s * 256], &sXn[s * BTILE * 2], buf, wave, lane);
    __syncthreads();
    sVl[s][rb][re] = sv[rb][re];
    float a = 0.f;
#pragma unroll
    for (int j = 0; j < 8; ++j) a += sv[rb][j] * buf[rb * 64 + j * 8 + re];
    sv[rb][re] = a;
  }
  __syncthreads();

  // Backward: Aright[site s] = sw before applying A_s ; sw <- A_s @ sw.
  for (int s = CHUNK - 1; s >= 0; --s) {
    float* buf = sA[s & 1];
    site_transfer_tile(&sW[s * 256], &sXn[s * BTILE * 2], buf, wave, lane);
    __syncthreads();
    sVr[s][rb][re] = sw[rb][re];
    float a = 0.f;
#pragma unroll
    for (int l = 0; l < 8; ++l) a += buf[rb * 64 + re * 8 + l] * sw[rb][l];
    sw[rb][re] = a;
  }
  __syncthreads();

  // Local H, apply to raw x, activation, L1 normalize.
  for (int p = t; p < CHUNK * BTILE; p += 128) {
    const int s = p / BTILE, b = p % BTILE;
    float H00 = 0.f, H01 = 0.f, H10 = 0.f, H11 = 0.f;
#pragma unroll
    for (int i = 0; i < 8; ++i) {
      const float vl = sVl[s][b][i];
#pragma unroll
      for (int l = 0; l < 8; ++l) {
        const float coef = vl * sVr[s][b][l];
        const float* Wp = &sW[s * 256 + i * 32 + l];
        H00 += coef * Wp[0];   // (j,k) = (0,0)
        H01 += coef * Wp[8];   // (0,1)
        H10 += coef * Wp[16];  // (1,0)
        H11 += coef * Wp[24];  // (1,1)
      }
    }
    const float x0 = sXr[p * 2 + 0], x1 = sXr[p * 2 + 1];
    float o0 = (H00 * x0 + H01 * x1) * (1.0f / NSITES);
    float o1 = (H10 * x0 + H11 * x1) * (1.0f / NSITES);
    o0 = o0 > 0.f ? o0 : -0.25f * o0;
    o1 = o1 > 0.f ? o1 : -0.25f * o1;
    const float a0 = fabsf(o0), a1 = fabsf(o1);
    const float inv = 1.0f / (a0 + a1);
    const size_t base = ((size_t)(c * CHUNK + s) * NBATCH + bt * BTILE + b) * 2;
    out[base + 0] = a0 * inv;
    out[base + 1] = a1 * inv;
  }
}

extern "C" void kernel_launch(void* const* d_in, const int* in_sizes, int n_in,
                              void* d_out, int out_size, void* d_ws,
                              size_t ws_size, hipStream_t stream)
{
  (void)in_sizes; (void)n_in; (void)out_size; (void)ws_size;
  const float* x   = (const float*)d_in[0];  // [N,B,d]
  const float* mpo = (const float*)d_in[1];  // [N,D,d,d,D]
  float* out = (float*)d_out;                // [N,B,d]

  float* P    = (float*)d_ws;                         // [C][B][8][8]  8 MB
  float* Lrow = P + (size_t)NCHUNK * NBATCH * 64;     // [C][B][8]     1 MB
  float* Rcol = Lrow + (size_t)NCHUNK * NBATCH * 8;   // [C][B][8]     1 MB

  dim3 grid(NCHUNK, NBATCH / BTILE);
  k_chunk_transfer<<<grid, 128, 0, stream>>>(x, mpo, P);
  k_boundary<<<(2 * NBATCH + 127) / 128, 128, 0, stream>>>(P, Lrow, Rcol);
  k_local<<<grid, 128, 0, stream>>>(x, mpo, Lrow, Rcol, out);
}